// BoundaryLoss_72344429134346
// MI455X (gfx1250) — compile-verified
//
#include <hip/hip_runtime.h>
#include <stdint.h>

// Problem dims (fixed by the reference).
#define B_    8
#define C_    4
#define H_    256
#define W_    256
#define ROWS  4                    // rows per pass-2 tile
#define BIGI  (H_ + W_)            // 512, reference sentinel
#define D2INIT 1048576.0f          // (2*BIG)^2, reference init
#define NPART ((C_ - 1) * B_ * (H_ / ROWS))   // 3*8*64 = 1536 partial sums

static_assert(NPART % 64 == 0, "final WMMA reduce assumes multiple of 64");

typedef __attribute__((ext_vector_type(2)))  float v2f;
typedef __attribute__((ext_vector_type(8)))  float v8f;
typedef __attribute__((ext_vector_type(4)))  unsigned int v4u;
typedef __attribute__((ext_vector_type(8)))  int v8i;
typedef __attribute__((ext_vector_type(4)))  int v4i;

// ---------------------------------------------------------------------------
// Pass 1: exact vertical 1-D distance (matches jax cummax formulation).
// One thread per (b, column w); loads coalesced across the wave (stride W
// between h iterations, consecutive w across lanes).
// Writes g^2 (float) to the ws scratch buffer.
// ---------------------------------------------------------------------------
__global__ void edt_vert(const int* __restrict__ mask,
                         float* __restrict__ g2, int cls)
{
    const int b = blockIdx.x;
    const int w = threadIdx.x;
    const int*  mcol = mask + (size_t)b * H_ * W_ + w;
    float*      gcol = g2   + (size_t)b * H_ * W_ + w;

    // down-scan: distance to nearest background above (bg = mask != cls)
    int last = -BIGI;                       // reference: last = -BIG
    for (int h = 0; h < H_; ++h) {
        if (mcol[h * W_] != cls) last = h;
        gcol[h * W_] = (float)(h - last);   // d_up
    }
    // up-scan: distance to nearest background below; combine, clamp, square
    int lastd = H_ - 1 + BIGI;              // flipped sentinel
    for (int h = H_ - 1; h >= 0; --h) {
        if (mcol[h * W_] != cls) lastd = h;
        float dup = gcol[h * W_];
        float dd  = (float)(lastd - h);
        float g   = fminf(fminf(dup, dd), (float)BIGI);
        gcol[h * W_] = g * g;
    }
}

// ---------------------------------------------------------------------------
// Pass 2: exact horizontal lower-envelope min, fused with pred*dist partial
// reduction. One block = ROWS rows of one batch image; g^2 tile staged in LDS
// via the Tensor Data Mover when available. Inner loop: broadcast LDS reads.
// ---------------------------------------------------------------------------
__global__ void __launch_bounds__(256)
edt_horiz_dot(const float* __restrict__ g2,
              const float* __restrict__ pred,
              float* __restrict__ partial, int cls)
{
    __shared__ float sg2[ROWS * W_];        // single shared symbol -> LDS offset 0

    const int hblk = blockIdx.x;            // 0..H/ROWS-1
    const int b    = blockIdx.y;            // 0..B-1
    const int h0   = hblk * ROWS;
    const int x    = threadIdx.x;           // 0..W-1

    const float* rowptr = g2 + ((size_t)b * H_ + h0) * W_;

#if defined(__gfx1250__) && __has_builtin(__builtin_amdgcn_tensor_load_to_lds) && \
    __has_builtin(__builtin_amdgcn_s_wait_tensorcnt)
    if (threadIdx.x < 32) {                 // one wave issues the DMA
        uint64_t ga = (uint64_t)rowptr;     // 57-bit byte address of tile start
        v4u g0;
        g0[0] = 1u;                                   // count=1, user descriptor
        g0[1] = 0u;                                   // lds_addr = 0 (sg2 is first)
        g0[2] = (unsigned int)ga;                     // global_addr[31:0]
        g0[3] = ((unsigned int)(ga >> 32) & 0x01FFFFFFu) | 0x80000000u; // addr[56:32] | type=2
        v8i g1;
        g1[0] = 0x00020000;                           // wg_mask=0, data_size=2 (4B)
        g1[1] = (int)((W_ & 0xFFFF) << 16);           // tensor_dim0 low16 (=256)
        g1[2] = (int)((W_ >> 16) | ((ROWS & 0xFFFF) << 16)); // dim0 hi | tensor_dim1 lo
        g1[3] = (int)((ROWS >> 16) | ((W_ & 0xFFFF) << 16)); // dim1 hi | tile_dim0
        g1[4] = (int)(ROWS & 0xFFFF);                 // tile_dim1 | tile_dim2=0
        g1[5] = (int)W_;                              // tensor_dim0_stride low32
        g1[6] = 0;                                    // stride0 hi | stride1 lo (unused, 2D)
        g1[7] = 0;
        v4i gz  = {0, 0, 0, 0};                       // groups 2/3 unused (2-D tensor)
        v8i gz8 = {0, 0, 0, 0, 0, 0, 0, 0};          // clang-23 6-arg form: extra group, zero
        __builtin_amdgcn_tensor_load_to_lds(g0, g1, gz, gz, gz8, 0);
        __builtin_amdgcn_s_wait_tensorcnt(0);
    }
    __syncthreads();
#else
    for (int i = threadIdx.x; i < ROWS * W_; i += 256) sg2[i] = rowptr[i];
    __syncthreads();
#endif

    const float* prow = pred + (((size_t)b * C_ + cls) * H_ + h0) * W_ + x;
    __builtin_prefetch(prow, 0, 0);         // global_prefetch for the pred rows

    float m[ROWS];
#pragma unroll
    for (int r = 0; r < ROWS; ++r) m[r] = D2INIT;

    // Exact brute-force lower envelope: D2[x] = min_xp g2[xp] + (x - xp)^2
    float df = (float)x;                    // df = x - xp, decremented each iter
#pragma unroll 4
    for (int xp = 0; xp < W_; ++xp) {
        float d2 = df * df;
#pragma unroll
        for (int r = 0; r < ROWS; ++r)
            m[r] = fminf(m[r], sg2[r * W_ + xp] + d2);
        df -= 1.0f;
    }

    const float inv_norm = 1.0f / (sqrtf((float)(H_ * H_ + W_ * W_)) + 1e-6f);
    float acc = 0.0f;
#pragma unroll
    for (int r = 0; r < ROWS; ++r) {
        float dist = sqrtf(m[r]) * inv_norm;
        acc += prow[(size_t)r * W_] * dist;
    }

    // Deterministic block reduction: shfl within wave32, LDS across 8 waves.
#pragma unroll
    for (int off = 16; off > 0; off >>= 1) acc += __shfl_xor(acc, off);
    __syncthreads();                        // done reading sg2; safe to reuse
    if ((threadIdx.x & 31) == 0) sg2[threadIdx.x >> 5] = acc;
    __syncthreads();
    if (threadIdx.x == 0) {
        float s = 0.0f;
#pragma unroll
        for (int i = 0; i < 8; ++i) s += sg2[i];
        partial[((size_t)(cls - 1) * B_ + b) * (H_ / ROWS) + hblk] = s;
    }
}

// ---------------------------------------------------------------------------
// Final reduction of NPART partials with V_WMMA_F32_16X16X4_F32 ("ones"
// trick): A holds 64 data values, B = ones, C accumulates row sums. Single
// wave, fixed iteration order -> bitwise deterministic output.
// ---------------------------------------------------------------------------
__global__ void __launch_bounds__(32)
final_reduce_wmma(const float* __restrict__ partial, float* __restrict__ out)
{
    const int lane = threadIdx.x;           // 32 lanes, EXEC all-ones for WMMA
    v2f ones; ones.x = 1.0f; ones.y = 1.0f;
    v8f acc = {};
    for (int i = 0; i < NPART; i += 64) {
        v2f a;
        a.x = partial[i + lane];
        a.y = partial[i + 32 + lane];
        acc = __builtin_amdgcn_wmma_f32_16x16x4_f32(
                  false, a, false, ones, (short)0, acc, false, false);
    }
    // lanes 0-15 hold rowsums M=0..7 in acc[0..7]; lanes 16-31 hold M=8..15.
    float s = acc[0] + acc[1] + acc[2] + acc[3] + acc[4] + acc[5] + acc[6] + acc[7];
    s += __shfl_xor(s, 16);                 // total now in every lane
    if (lane == 0) {
        const float scale = 1.0f / ((float)B_ * H_ * W_ * (C_ - 1));
        out[0] = s * scale;
    }
}

// ---------------------------------------------------------------------------
extern "C" void kernel_launch(void* const* d_in, const int* in_sizes, int n_in,
                              void* d_out, int out_size, void* d_ws, size_t ws_size,
                              hipStream_t stream)
{
    const float* pred = (const float*)d_in[0];   // [B,C,H,W] f32 softmax
    const int*   mask = (const int*)d_in[1];     // [B,H,W] int
    float* out = (float*)d_out;                  // scalar

    float* g2      = (float*)d_ws;                       // B*H*W floats (2 MB)
    float* partial = g2 + (size_t)B_ * H_ * W_;          // NPART floats

    for (int cls = 1; cls < C_; ++cls) {
        edt_vert<<<dim3(B_), dim3(W_), 0, stream>>>(mask, g2, cls);
        edt_horiz_dot<<<dim3(H_ / ROWS, B_), dim3(256), 0, stream>>>(g2, pred, partial, cls);
    }
    final_reduce_wmma<<<1, 32, 0, stream>>>(partial, out);
}